// FixedMultiHeadGraphAttention2D_7413113553201
// MI455X (gfx1250) — compile-verified
//
#include <hip/hip_runtime.h>

typedef float v2f __attribute__((ext_vector_type(2)));
typedef float v8f __attribute__((ext_vector_type(8)));

#define IN_DIM 128
#define OUT_DIM 128
#define NHEAD 8
#define HDIM 16
#define LN_EPS 1e-5f
#define LDS_PITCH 132   // 128 + 4 pad floats -> bank-conflict-free A reads

// ---------- float atomic-max via order-preserving int encoding ----------
__device__ __forceinline__ int enc_f(float x) {
    int i = __float_as_int(x);
    return i >= 0 ? i : (i ^ 0x7FFFFFFF);
}
__device__ __forceinline__ float dec_f(int i) {
    return __int_as_float(i >= 0 ? i : (i ^ 0x7FFFFFFF));
}

// ---------- init workspace ----------
__global__ __launch_bounds__(256) void init_kernel(float* __restrict__ AGG,
                                                   float* __restrict__ DEN,
                                                   int* __restrict__ MAXE,
                                                   int nAgg, int nDen) {
    int i = blockIdx.x * blockDim.x + threadIdx.x;
    if (i < nAgg) AGG[i] = 0.0f;
    if (i < nDen) { DEN[i] = 0.0f; MAXE[i] = (int)0x80000000; }
}

// ---------- fused Q/K/V/Skip projection GEMM (WMMA f32 16x16x4) ----------
// One block per 16-row tile. The 16x128 A tile of x is staged into LDS ONCE
// via the CDNA5 async-to-LDS DMA path (ASYNCcnt), then each of the 8 waves
// computes its 16-column slice for all four projections.
// weight layout: W[h][i][d] -> element (k, col=h*16+d) at h*2048 + k*16 + d
__global__ __launch_bounds__(256) void proj_kernel(
    const float* __restrict__ x,
    const float* __restrict__ Wq, const float* __restrict__ bq,
    const float* __restrict__ Wk, const float* __restrict__ bk,
    const float* __restrict__ Wv, const float* __restrict__ bv,
    const float* __restrict__ Wsk, const float* __restrict__ bsk,
    float* __restrict__ Q, float* __restrict__ K,
    float* __restrict__ V, float* __restrict__ S, int N)
{
    __shared__ float lds[16 * LDS_PITCH];
    const int rbase = blockIdx.x * 16;

    // async stage: 16 rows x 128 floats = 512 x b128 chunks / 16B each,
    // 2 chunks per thread (wave w copies rows w and w+8, 16B per lane).
    #pragma unroll
    for (int i = threadIdx.x; i < 512; i += 256) {
        int r = i >> 5;
        int c = (i & 31) << 2;
        int rr = rbase + r; rr = rr < N ? rr : N - 1;
        unsigned long long g =
            (unsigned long long)(uintptr_t)(x + (size_t)rr * IN_DIM + c);
        unsigned int l = (unsigned int)(uintptr_t)&lds[r * LDS_PITCH + c];
        asm volatile("global_load_async_to_lds_b128 %0, %1, off"
                     :: "v"(l), "v"(g) : "memory");
    }
    asm volatile("s_wait_asynccnt 0x0" ::: "memory");
    __syncthreads();

    const int wave = threadIdx.x >> 5;
    const int lane = threadIdx.x & 31;
    const int m  = lane & 15;              // A row / B,D column within tile
    const int kh = (lane >> 4) << 1;       // K sub-block: 0 or 2
    const int mo = (lane >> 4) << 3;       // D row offset: 0 or 8
    const int cb = wave << 4;              // this wave's column base
    const int col = cb + m;                // h = wave, d = m

    #pragma unroll 1
    for (int tt = 0; tt < 4; ++tt) {
        const float* W; const float* bias; float* O;
        if      (tt == 0) { W = Wq;  bias = bq;  O = Q; }
        else if (tt == 1) { W = Wk;  bias = bk;  O = K; }
        else if (tt == 2) { W = Wv;  bias = bv;  O = V; }
        else              { W = Wsk; bias = bsk; O = S; }
        const float* Wc = W + (wave << 11) + m;   // W'(k, col) = Wc[k*16]

        v8f acc = {};
        #pragma unroll 8
        for (int k0 = 0; k0 < IN_DIM; k0 += 4) {
            v2f a = *(const v2f*)&lds[m * LDS_PITCH + k0 + kh];
            v2f b; b.x = Wc[(k0 + kh) << 4]; b.y = Wc[(k0 + kh + 1) << 4];
            acc = __builtin_amdgcn_wmma_f32_16x16x4_f32(
                false, a, false, b, (short)0, acc, false, false);
        }

        const float bb = bias[col];
        if (rbase + 16 <= N) {             // uniform fast path: no exec churn
            #pragma unroll
            for (int r = 0; r < 8; ++r)
                O[(size_t)(rbase + mo + r) * OUT_DIM + col] = acc[r] + bb;
        } else {
            #pragma unroll
            for (int r = 0; r < 8; ++r) {
                int orow = rbase + mo + r;
                if (orow < N) O[(size_t)orow * OUT_DIM + col] = acc[r] + bb;
            }
        }
    }
}

// ---------- edge pass 1: attention logits + segment max ----------
__global__ __launch_bounds__(256) void edge_pass1(
    const int* __restrict__ ei, const float* __restrict__ ea,
    const float* __restrict__ Q, const float* __restrict__ K,
    const float* __restrict__ We,
    float* __restrict__ ALPHA, int* __restrict__ MAXE, int E)
{
    const int wave = threadIdx.x >> 5;
    const int lane = threadIdx.x & 31;
    const int e = blockIdx.x * 8 + wave;
    if (e >= E) return;

    const int src = ei[e];
    const int dst = ei[E + e];
    const float w = ea[e];
    const float* qrow = Q + (size_t)dst * OUT_DIM;
    const float* krow = K + (size_t)src * OUT_DIM;

    float part[4];
    #pragma unroll
    for (int j = 0; j < 4; ++j) {
        int c = j * 32 + lane;
        part[j] = qrow[c] * (krow[c] + w * We[c]);
    }
    // segmented reduction within 16-lane halves
    #pragma unroll
    for (int msk = 1; msk < 16; msk <<= 1) {
        #pragma unroll
        for (int j = 0; j < 4; ++j) part[j] += __shfl_xor(part[j], msk, 32);
    }
    if ((lane & 15) == 0) {
        const int hb = lane >> 4;                // 0 or 1
        #pragma unroll
        for (int j = 0; j < 4; ++j) {
            int h = 2 * j + hb;
            float al = part[j] * 0.25f;          // 1/sqrt(16)
            ALPHA[(size_t)e * NHEAD + h] = al;
            atomicMax(&MAXE[(size_t)dst * NHEAD + h], enc_f(al));
        }
    }
}

// ---------- edge pass 2: exp, denom, unnormalized weighted aggregation ----------
__global__ __launch_bounds__(256) void edge_pass2(
    const int* __restrict__ ei, const float* __restrict__ ea,
    const float* __restrict__ V, const float* __restrict__ We,
    const float* __restrict__ ALPHA, const int* __restrict__ MAXE,
    float* __restrict__ DEN, float* __restrict__ AGG, int E)
{
    const int wave = threadIdx.x >> 5;
    const int lane = threadIdx.x & 31;
    const int e = blockIdx.x * 8 + wave;
    if (e >= E) return;

    const int src = ei[e];
    const int dst = ei[E + e];
    const float w = ea[e];

    float exv = 0.0f;
    if (lane < NHEAD) {
        float al = ALPHA[(size_t)e * NHEAD + lane];
        float mv = dec_f(MAXE[(size_t)dst * NHEAD + lane]);
        exv = __expf(al - mv);
        atomicAdd(&DEN[(size_t)dst * NHEAD + lane], exv);
    }
    const float* vrow = V + (size_t)src * OUT_DIM;
    #pragma unroll
    for (int j = 0; j < 4; ++j) {
        int c = j * 32 + lane;
        int h = 2 * j + (lane >> 4);
        float exh = __shfl(exv, h, 32);
        float val = exh * (vrow[c] + w * We[c]);
        atomicAdd(&AGG[(size_t)dst * OUT_DIM + c], val);
    }
}

// ---------- normalize aggregation + skip (in place on AGG) ----------
__global__ __launch_bounds__(256) void hc_kernel(
    float* __restrict__ AGG, const float* __restrict__ DEN,
    const float* __restrict__ S, int total)
{
    int i = blockIdx.x * blockDim.x + threadIdx.x;
    if (i >= total) return;
    int n = i >> 7;
    int h = (i & 127) >> 4;
    float den = DEN[(size_t)n * NHEAD + h];
    AGG[i] = AGG[i] / fmaxf(den, 1e-16f) + S[i];
}

// ---------- output GEMM (WMMA f32) + bias + residual ----------
__global__ __launch_bounds__(256) void outgemm_kernel(
    const float* __restrict__ HC, const float* __restrict__ Wo,
    const float* __restrict__ bo, const float* __restrict__ x,
    float* __restrict__ T, int N)
{
    __shared__ float lds[16 * LDS_PITCH];
    const int rbase = blockIdx.x * 16;

    #pragma unroll
    for (int i = threadIdx.x; i < 512; i += 256) {
        int r = i >> 5;
        int c = (i & 31) << 2;
        int rr = rbase + r; rr = rr < N ? rr : N - 1;
        unsigned long long g =
            (unsigned long long)(uintptr_t)(HC + (size_t)rr * OUT_DIM + c);
        unsigned int l = (unsigned int)(uintptr_t)&lds[r * LDS_PITCH + c];
        asm volatile("global_load_async_to_lds_b128 %0, %1, off"
                     :: "v"(l), "v"(g) : "memory");
    }
    asm volatile("s_wait_asynccnt 0x0" ::: "memory");
    __syncthreads();

    const int wave = threadIdx.x >> 5;
    const int lane = threadIdx.x & 31;
    const int m  = lane & 15;
    const int kh = (lane >> 4) << 1;
    const int mo = (lane >> 4) << 3;
    const int col = (wave << 4) + m;

    v8f acc = {};
    #pragma unroll 8
    for (int k0 = 0; k0 < OUT_DIM; k0 += 4) {
        v2f a = *(const v2f*)&lds[m * LDS_PITCH + k0 + kh];
        v2f b; b.x = Wo[(k0 + kh) * OUT_DIM + col];
               b.y = Wo[(k0 + kh + 1) * OUT_DIM + col];
        acc = __builtin_amdgcn_wmma_f32_16x16x4_f32(
            false, a, false, b, (short)0, acc, false, false);
    }

    const float bb = bo[col];
    if (rbase + 16 <= N) {
        #pragma unroll
        for (int r = 0; r < 8; ++r) {
            int orow = rbase + mo + r;
            T[(size_t)orow * OUT_DIM + col] =
                acc[r] + bb + x[(size_t)orow * OUT_DIM + col];
        }
    } else {
        #pragma unroll
        for (int r = 0; r < 8; ++r) {
            int orow = rbase + mo + r;
            if (orow < N)
                T[(size_t)orow * OUT_DIM + col] =
                    acc[r] + bb + x[(size_t)orow * OUT_DIM + col];
        }
    }
}

// ---------- LayerNorm, wave per node, in place ----------
__global__ __launch_bounds__(256) void ln_kernel(
    float* __restrict__ out, const float* __restrict__ gamma,
    const float* __restrict__ beta, int N)
{
    const int wave = threadIdx.x >> 5;
    const int lane = threadIdx.x & 31;
    const int n = blockIdx.x * 8 + wave;
    if (n >= N) return;

    float t[4], s = 0.0f, s2 = 0.0f;
    #pragma unroll
    for (int j = 0; j < 4; ++j) {
        int c = j * 32 + lane;
        t[j] = out[(size_t)n * OUT_DIM + c];
        s += t[j]; s2 += t[j] * t[j];
    }
    #pragma unroll
    for (int msk = 1; msk < 32; msk <<= 1) {
        s  += __shfl_xor(s,  msk, 32);
        s2 += __shfl_xor(s2, msk, 32);
    }
    const float mu  = s * (1.0f / OUT_DIM);
    const float var = s2 * (1.0f / OUT_DIM) - mu * mu;
    const float inv = rsqrtf(var + LN_EPS);
    #pragma unroll
    for (int j = 0; j < 4; ++j) {
        int c = j * 32 + lane;
        out[(size_t)n * OUT_DIM + c] = (t[j] - mu) * inv * gamma[c] + beta[c];
    }
}

extern "C" void kernel_launch(void* const* d_in, const int* in_sizes, int n_in,
                              void* d_out, int out_size, void* d_ws, size_t ws_size,
                              hipStream_t stream) {
    const float* x   = (const float*)d_in[0];
    const int*   ei  = (const int*)  d_in[1];
    const float* ea  = (const float*)d_in[2];
    const float* Wq  = (const float*)d_in[3];
    const float* bq  = (const float*)d_in[4];
    const float* Wk  = (const float*)d_in[5];
    const float* bk  = (const float*)d_in[6];
    const float* Wv  = (const float*)d_in[7];
    const float* bv  = (const float*)d_in[8];
    const float* We  = (const float*)d_in[9];
    const float* Wsk = (const float*)d_in[10];
    const float* bsk = (const float*)d_in[11];
    const float* Wo  = (const float*)d_in[12];
    const float* bo  = (const float*)d_in[13];
    const float* gam = (const float*)d_in[14];
    const float* bet = (const float*)d_in[15];
    float* out = (float*)d_out;

    const int N = in_sizes[0] / IN_DIM;
    const int E = in_sizes[2];

    float* ws = (float*)d_ws;
    const size_t NF = (size_t)N * OUT_DIM;
    float* Q     = ws;
    float* K     = Q + NF;
    float* V     = K + NF;
    float* S     = V + NF;
    float* AGG   = S + NF;
    float* ALPHA = AGG + NF;                 // E*8 floats
    float* DEN   = ALPHA + (size_t)E * NHEAD;
    int*   MAXE  = (int*)(DEN + (size_t)N * NHEAD);

    const int nAgg = N * OUT_DIM;
    const int nDen = N * NHEAD;

    init_kernel<<<(nAgg + 255) / 256, 256, 0, stream>>>(AGG, DEN, MAXE, nAgg, nDen);

    proj_kernel<<<(N + 15) / 16, 256, 0, stream>>>(x, Wq, bq, Wk, bk, Wv, bv,
                                                   Wsk, bsk, Q, K, V, S, N);

    edge_pass1<<<(E + 7) / 8, 256, 0, stream>>>(ei, ea, Q, K, We, ALPHA, MAXE, E);
    edge_pass2<<<(E + 7) / 8, 256, 0, stream>>>(ei, ea, V, We, ALPHA, MAXE, DEN, AGG, E);

    hc_kernel<<<(nAgg + 255) / 256, 256, 0, stream>>>(AGG, DEN, S, nAgg);

    outgemm_kernel<<<(N + 15) / 16, 256, 0, stream>>>(AGG, Wo, bo, x, out, N);

    ln_kernel<<<(N + 7) / 8, 256, 0, stream>>>(out, gam, bet, N);
}